// GNN_46273977647663
// MI455X (gfx1250) — compile-verified
//
#include <hip/hip_runtime.h>

#define N_NODES 100000
#define N_EDGES 1000000
#define D 64
#define NLAYERS 4
#define N_GRAPHS 1024
#define OUT_DIM 16
#define KP_STRIDE 160  // padded dwords per k-pair row: khalf halves land in disjoint bank windows

typedef __attribute__((ext_vector_type(2))) float v2f;
typedef __attribute__((ext_vector_type(8))) float v8f;

// f32 WMMA: D(16x16,f32) = A(16x4,f32) * B(4x16,f32) + C. K=4 per issue.
__device__ __forceinline__ v8f wmma4_f32(v2f a, v2f b, v8f c) {
#if defined(__gfx1250__)
    return __builtin_amdgcn_wmma_f32_16x16x4_f32(
        /*neg_a=*/false, a, /*neg_b=*/false, b,
        /*c_mod=*/(short)0, c, /*reuse_a=*/false, /*reuse_b=*/false);
#else
    (void)a; (void)b;
    return c;
#endif
}

__global__ void k_zero(float* __restrict__ p, int n4) {
    int i = blockIdx.x * blockDim.x + threadIdx.x;
    if (i < n4) ((float4*)p)[i] = make_float4(0.f, 0.f, 0.f, 0.f);
}

// ---------------- CSR build (once per launch) ----------------

// deg[dst]++ for every edge (int atomics, cheap).
__global__ void k_count(const int* __restrict__ ei, int* __restrict__ deg) {
    int e = blockIdx.x * blockDim.x + threadIdx.x;
    if (e < N_EDGES) atomicAdd(&deg[ei[N_EDGES + e]], 1);
}

// Single-block exclusive scan of deg -> start; re-init deg as fill cursor.
// deg_cursor is read (degree) then overwritten (cursor) element-by-element.
__global__ __launch_bounds__(1024) void k_scan(int* deg_cursor, int* start) {
    __shared__ int part[1024];
    const int t = threadIdx.x;
    const int CH = (N_NODES + 1023) / 1024;
    const int lo = t * CH;
    const int hi = (lo + CH < N_NODES) ? lo + CH : N_NODES;
    int s = 0;
    for (int i = lo; i < hi; ++i) s += deg_cursor[i];
    part[t] = s;
    __syncthreads();
    if (t == 0) {
        int run = 0;
        for (int i = 0; i < 1024; ++i) { int v = part[i]; part[i] = run; run += v; }
    }
    __syncthreads();
    int run = part[t];
    for (int i = lo; i < hi; ++i) {
        int v = deg_cursor[i];
        start[i] = run;
        deg_cursor[i] = run;  // cursor for k_fill
        run += v;
    }
    if (t == 1023) start[N_NODES] = run;  // == N_EDGES
}

// Bucket edge sources by destination.
__global__ void k_fill(const int* __restrict__ ei, int* cursor, int* __restrict__ esrc) {
    int e = blockIdx.x * blockDim.x + threadIdx.x;
    if (e < N_EDGES) {
        int d = ei[N_EDGES + e];
        int pos = atomicAdd(&cursor[d], 1);
        esrc[pos] = ei[e];
    }
}

// ---------------- per-layer atomic-free aggregation ----------------

// One wave per node: lane owns 2 columns, accumulate h[src] rows in registers.
__global__ __launch_bounds__(256) void k_gather(const float* __restrict__ h,
                                                const int* __restrict__ start,
                                                const int* __restrict__ esrc,
                                                float* __restrict__ agg) {
    const int wave = threadIdx.x >> 5;
    const int lane = threadIdx.x & 31;
    const int node = blockIdx.x * 8 + wave;
    if (node >= N_NODES) return;
    const int e0 = start[node];
    const int e1 = start[node + 1];
    const int c = lane * 2;
    v2f acc = {0.f, 0.f};
    for (int e = e0; e < e1; ++e) {
        const int s = esrc[e];
        acc += *(const v2f*)&h[(long)s * D + c];
    }
    *(v2f*)&agg[(long)node * D + c] = acc;
}

// ---------------- WMMA layer: hout = relu(agg@Wrel + brel + hin@Wroot) ----------------

__global__ __launch_bounds__(256) void k_layer(
    const float* __restrict__ agg, const float* __restrict__ hin,
    float* __restrict__ hout, const float* __restrict__ Wrel,
    const float* __restrict__ brel, const float* __restrict__ Wroot) {
    // k-pair interleaved weight staging: sW[(k>>1)*KP_STRIDE + n*2 + (k&1)]
    // -> B fragment {W[k][n], W[k+1][n]} is one aligned ds_load_b64.
    __shared__ float sWrel[32 * KP_STRIDE];
    __shared__ float sWroot[32 * KP_STRIDE];
    const int tid = threadIdx.x;
    for (int i = tid; i < D * D; i += 256) {
        const int k = i >> 6, n = i & 63;
        const int idx = (k >> 1) * KP_STRIDE + n * 2 + (k & 1);
        sWrel[idx]  = Wrel[i];
        sWroot[idx] = Wroot[i];
    }
    __syncthreads();

    const int wave  = tid >> 5;
    const int lane  = tid & 31;
    const int row0  = blockIdx.x * 128 + wave * 16;
    if (row0 >= N_NODES) return;           // wave-uniform: EXEC stays all-1s

    const int mrow  = lane & 15;
    const int khalf = lane >> 4;
    const int col   = lane & 15;
    const long arow = (long)(row0 + mrow) * D;

    const v8f z = {0.f, 0.f, 0.f, 0.f, 0.f, 0.f, 0.f, 0.f};
    v8f acc[4] = {z, z, z, z};

#pragma unroll
    for (int kk = 0; kk < 16; ++kk) {
        const int kb = kk * 4 + khalf * 2;
        const int p  = kk * 2 + khalf;     // k-pair row in staged weights
        const v2f aR = *(const v2f*)&agg[arow + kb];
        const v2f aH = *(const v2f*)&hin[arow + kb];
#pragma unroll
        for (int nt = 0; nt < 4; ++nt) {
            const int bi = p * KP_STRIDE + (nt * 16 + col) * 2;
            const v2f bR = *(const v2f*)&sWrel[bi];
            const v2f bH = *(const v2f*)&sWroot[bi];
            acc[nt] = wmma4_f32(aR, bR, acc[nt]);
            acc[nt] = wmma4_f32(aH, bH, acc[nt]);
        }
    }

#pragma unroll
    for (int nt = 0; nt < 4; ++nt) {
        const float bias = brel[nt * 16 + col];
#pragma unroll
        for (int r = 0; r < 8; ++r) {
            const int orow = row0 + r + khalf * 8;
            float v = acc[nt][r] + bias;
            hout[(long)orow * D + nt * 16 + col] = v > 0.f ? v : 0.f;
        }
    }
}

// ---------------- pooling + head ----------------

__global__ void k_pool(const float* __restrict__ h, const int* __restrict__ batch,
                       float* __restrict__ g) {
    int tid = blockIdx.x * blockDim.x + threadIdx.x;
    if (tid >= N_NODES * 16) return;
    int node = tid >> 4;
    int t = (tid & 15) * 4;
    int b = batch[node];
    const float4 v = *(const float4*)&h[(long)node * D + t];
    float* p = &g[(long)b * D + t];
    atomicAdd(p + 0, v.x);
    atomicAdd(p + 1, v.y);
    atomicAdd(p + 2, v.z);
    atomicAdd(p + 3, v.w);
}

// y = relu(g @ W1 + b1) @ W2 + b2 ; one wave handles 16 graphs.
__global__ __launch_bounds__(32) void k_head(
    const float* __restrict__ g, const float* __restrict__ W1,
    const float* __restrict__ b1, const float* __restrict__ W2,
    const float* __restrict__ b2, float* __restrict__ out) {
    __shared__ float sW1[32 * KP_STRIDE];
    __shared__ float sW2[32 * 32];         // (k>>1)*32 + n*2 + (k&1), n<16
    __shared__ float hid[16 * D];
    const int lane = threadIdx.x;
    for (int i = lane; i < D * D; i += 32) {
        const int k = i >> 6, n = i & 63;
        sW1[(k >> 1) * KP_STRIDE + n * 2 + (k & 1)] = W1[i];
    }
    for (int i = lane; i < D * OUT_DIM; i += 32) {
        const int k = i >> 4, n = i & 15;
        sW2[(k >> 1) * 32 + n * 2 + (k & 1)] = W2[i];
    }
    __syncthreads();

    const int row0  = blockIdx.x * 16;
    const int mrow  = lane & 15;
    const int khalf = lane >> 4;
    const int col   = lane & 15;
    const long arow = (long)(row0 + mrow) * D;

    const v8f z = {0.f, 0.f, 0.f, 0.f, 0.f, 0.f, 0.f, 0.f};
    v8f acc[4] = {z, z, z, z};
#pragma unroll
    for (int kk = 0; kk < 16; ++kk) {
        const int kb = kk * 4 + khalf * 2;
        const int p  = kk * 2 + khalf;
        const v2f a = *(const v2f*)&g[arow + kb];
#pragma unroll
        for (int nt = 0; nt < 4; ++nt) {
            const v2f b = *(const v2f*)&sW1[p * KP_STRIDE + (nt * 16 + col) * 2];
            acc[nt] = wmma4_f32(a, b, acc[nt]);
        }
    }
#pragma unroll
    for (int nt = 0; nt < 4; ++nt) {
        const float bias = b1[nt * 16 + col];
#pragma unroll
        for (int r = 0; r < 8; ++r) {
            float v = acc[nt][r] + bias;
            hid[(r + khalf * 8) * D + nt * 16 + col] = v > 0.f ? v : 0.f;
        }
    }
    __syncthreads();  // single wave: orders LDS stores vs loads

    v8f c2 = z;
#pragma unroll
    for (int kk = 0; kk < 16; ++kk) {
        const int kb = kk * 4 + khalf * 2;
        const int p  = kk * 2 + khalf;
        const v2f a = *(const v2f*)&hid[mrow * D + kb];
        const v2f b = *(const v2f*)&sW2[p * 32 + col * 2];
        c2 = wmma4_f32(a, b, c2);
    }
    const float bias2 = b2[col];
#pragma unroll
    for (int r = 0; r < 8; ++r)
        out[(row0 + r + khalf * 8) * OUT_DIM + col] = c2[r] + bias2;
}

extern "C" void kernel_launch(void* const* d_in, const int* in_sizes, int n_in,
                              void* d_out, int out_size, void* d_ws, size_t ws_size,
                              hipStream_t stream) {
    const float* x     = (const float*)d_in[0];
    const int*   ei    = (const int*)d_in[1];
    const int*   batch = (const int*)d_in[2];
    const float* Wrel  = (const float*)d_in[3];
    const float* brel  = (const float*)d_in[4];
    const float* Wroot = (const float*)d_in[5];
    const float* W1    = (const float*)d_in[6];
    const float* b1    = (const float*)d_in[7];
    const float* W2    = (const float*)d_in[8];
    const float* b2    = (const float*)d_in[9];
    float* out = (float*)d_out;

    // workspace layout
    float* agg    = (float*)d_ws;                         // N*D
    float* hbuf   = agg + (size_t)N_NODES * D;            // N*D
    float* g      = hbuf + (size_t)N_NODES * D;           // G*D
    int*   start  = (int*)(g + (size_t)N_GRAPHS * D);     // N+1
    int*   cursor = start + (N_NODES + 1);                // N (deg, then fill cursor)
    int*   esrc   = cursor + N_NODES;                     // E

    // ---- CSR build by destination (atomic-free per-layer aggregation) ----
    k_zero<<<(N_NODES / 4 + 255) / 256, 256, 0, stream>>>((float*)cursor, N_NODES / 4);
    k_count<<<(N_EDGES + 255) / 256, 256, 0, stream>>>(ei, cursor);
    k_scan<<<1, 1024, 0, stream>>>(cursor, start);
    k_fill<<<(N_EDGES + 255) / 256, 256, 0, stream>>>(ei, cursor, esrc);

    // ---- layers ----
    for (int l = 0; l < NLAYERS; ++l) {
        const float* hin = (l == 0) ? x : hbuf;
        k_gather<<<(N_NODES + 7) / 8, 256, 0, stream>>>(hin, start, esrc, agg);
        k_layer<<<(N_NODES + 127) / 128, 256, 0, stream>>>(
            agg, hin, hbuf, Wrel + (size_t)l * D * D, brel + (size_t)l * D,
            Wroot + (size_t)l * D * D);
    }

    // ---- pooling + head ----
    const int GD4 = N_GRAPHS * D / 4;
    k_zero<<<(GD4 + 255) / 256, 256, 0, stream>>>(g, GD4);
    k_pool<<<(N_NODES * 16 + 255) / 256, 256, 0, stream>>>(hbuf, batch, g);
    k_head<<<N_GRAPHS / 16, 32, 0, stream>>>(g, W1, b1, W2, b2, out);
}